// Decoder_91190745628725
// MI455X (gfx1250) — compile-verified
//
#include <hip/hip_runtime.h>
#include <math.h>

#define D 1024
#define T 8192
#define NWG 64
#define ROWS_PER_WG (D / NWG)            /* 16 */
#define SCAN_THREADS 256
#define WAVES (SCAN_THREADS / 32)        /* 8 waves (wave32) */
#define ROWS_PER_WAVE (ROWS_PER_WG / WAVES) /* 2 */

typedef float v2f __attribute__((ext_vector_type(2)));
typedef float v8f __attribute__((ext_vector_type(8)));
typedef unsigned int u32x4 __attribute__((ext_vector_type(4)));
typedef int i32x4 __attribute__((ext_vector_type(4)));
typedef int i32x8 __attribute__((ext_vector_type(8)));

#if defined(__has_builtin)
#if __has_builtin(__builtin_amdgcn_tensor_load_to_lds) && \
    __has_builtin(__builtin_amdgcn_s_wait_tensorcnt)
#define HAVE_TDM 1
#endif
#endif

__device__ __forceinline__ float sigmoidf_(float x) {
    return 1.0f / (1.0f + __expf(-x));
}

// CDNA5 async LDS fill: GLOBAL_LOAD_ASYNC_TO_LDS_B128 (tracked by ASYNCcnt).
// VDST vgpr carries the LDS byte offset (low 32 bits of a generic shared ptr).
__device__ __forceinline__ void async_to_lds_b128(void* lds_dst, const void* gsrc) {
    unsigned ldst = (unsigned)(uintptr_t)lds_dst;           // LDS byte offset
    unsigned long long ga = (unsigned long long)(uintptr_t)gsrc;
    asm volatile("global_load_async_to_lds_b128 %0, %1, off"
                 :: "v"(ldst), "v"(ga) : "memory");
}
__device__ __forceinline__ void wait_async0() {
    asm volatile("s_wait_asynccnt 0x0" ::: "memory");
}

// ---------------------------------------------------------------------------
// Init: copy encoder state into double buffers, zero the grid-barrier counter.
// Runs every kernel_launch -> deterministic across graph replays.
// ---------------------------------------------------------------------------
__global__ void init_state(const float* __restrict__ h0, const float* __restrict__ c0,
                           float* __restrict__ Hb, float* __restrict__ Cb,
                           unsigned* __restrict__ bar) {
    int i = blockIdx.x * blockDim.x + threadIdx.x;
    if (i < D) { Hb[i] = h0[i]; Cb[i] = c0[i]; }
    if (i < 64) bar[i] = 0u;
}

// ---------------------------------------------------------------------------
// Pack the x-side halves of W_f/W_i/W_o into k-major (K x N) layout so the
// WMMA B-operand loads are coalesced: Wt[g][k][n] = W_g[n][D + k].
// ---------------------------------------------------------------------------
__global__ void pack_wx(const float* __restrict__ Wf, const float* __restrict__ Wi,
                        const float* __restrict__ Wo, float* __restrict__ Wt) {
    int idx = blockIdx.x * blockDim.x + threadIdx.x;   // over 3*D*D
    int g   = idx >> 20;                               // D*D == 1<<20
    int rem = idx & ((1 << 20) - 1);
    int k = rem >> 10;
    int n = rem & (D - 1);
    const float* W = (g == 0) ? Wf : (g == 1) ? Wi : Wo;
    Wt[idx] = W[(size_t)n * (2 * D) + D + k];
}

// ---------------------------------------------------------------------------
// Parallel phase: Xp[g][t][n] = sum_k X[t][k] * Wt[g][k][n]
// f32 WMMA 16x16x4. Each wave computes TWO adjacent 16x16 tiles (reuses the A
// fragment, 2 wmma per K-step); a block of 8 waves shares one 16-row A panel
// staged in LDS via async b128 loads (64 KB dynamic).
//
// VGPR layouts (CDNA5 ISA 7.12.2):
//   A 16x4 f32 : m=lane%16, v0 -> K=koff, v1 -> K=koff+1, koff=2*(lane/16).
//   B 4x16 f32 : N=lane%16, same K split mirrored across lane halves.
//   C/D 16x16  : vgpr j -> M = j + 8*(lane/16), N = lane%16.
// ---------------------------------------------------------------------------
__global__ void xproj_wmma(const float* __restrict__ X, const float* __restrict__ Wt,
                           float* __restrict__ Xp) {
    extern __shared__ float As[];                      // 16 x D floats
    const int g = blockIdx.z;
    const float* Wg = Wt + (size_t)g * D * D;
    float* Pg = Xp + (size_t)g * T * D;
    const int t0 = blockIdx.y * 16;
    const int tid = threadIdx.x;

    // Async-stage the 16 x 1024 A panel: 4096 float4s, 16 per thread.
    for (int e4 = tid; e4 < 16 * D / 4; e4 += blockDim.x) {
        const float* gsrc = X + (size_t)(t0 + (e4 >> 8)) * D + (e4 & 255) * 4;
        async_to_lds_b128(As + e4 * 4, gsrc);
    }
    wait_async0();
    __syncthreads();

    const int lane = tid & 31;
    const int wave = tid >> 5;
    const int n0   = (blockIdx.x * (WAVES * 2) + wave * 2) * 16;
    const int m    = lane & 15;
    const int koff = (lane >> 4) << 1;                 // 0 or 2

    v8f acc0 = {}, acc1 = {};
#pragma unroll 8
    for (int k0 = 0; k0 < D; k0 += 4) {
        v2f a, b0, b1;
        a.x  = As[m * D + k0 + koff];
        a.y  = As[m * D + k0 + koff + 1];
        const float* brow0 = Wg + (size_t)(k0 + koff) * D + n0 + m;
        const float* brow1 = Wg + (size_t)(k0 + koff + 1) * D + n0 + m;
        b0.x = brow0[0];   b0.y = brow1[0];
        b1.x = brow0[16];  b1.y = brow1[16];
        acc0 = __builtin_amdgcn_wmma_f32_16x16x4_f32(false, a, false, b0,
                                                     (short)0, acc0, false, false);
        acc1 = __builtin_amdgcn_wmma_f32_16x16x4_f32(false, a, false, b1,
                                                     (short)0, acc1, false, false);
    }

    const int mbase = (lane >> 4) << 3;                // 0 or 8
#pragma unroll
    for (int j = 0; j < 8; ++j) {
        Pg[(size_t)(t0 + mbase + j) * D + n0 + m]      = acc0[j];
        Pg[(size_t)(t0 + mbase + j) * D + n0 + 16 + m] = acc1[j];
    }
}

// ---------------------------------------------------------------------------
// Sequential phase: persistent cooperative kernel, 64 WGs x 256 threads.
// Each WG owns 16 state rows; h-side rows of W_f/W_i/W_o/W_C live in LDS
// (256 KB, async-filled once). Per step, wave 0 issues ONE Tensor Data Mover
// descriptor that DMAs h_{t-1} and c_{t-1} (2 rows x 4 KB, row stride 8 KB)
// into LDS, then s_wait_tensorcnt 0. One grid barrier per step; cluster HW
// barrier fast path when launched in a cluster.
// ---------------------------------------------------------------------------
__global__ void __launch_bounds__(SCAN_THREADS, 1)
lstm_scan(const float* __restrict__ Wf, const float* __restrict__ Wi,
          const float* __restrict__ Wo, const float* __restrict__ WC,
          const float* __restrict__ bf, const float* __restrict__ bi,
          const float* __restrict__ bC, const float* __restrict__ bo,
          const float* __restrict__ Xp, float* __restrict__ Hb,
          float* __restrict__ Cb, unsigned* __restrict__ bar,
          float* __restrict__ out) {
    extern __shared__ float smem[];
    const int MS = ROWS_PER_WG * D;                    // 16384 floats / matrix
    float* wlds = smem;                                // 4*MS floats (256 KB)
    float* hld  = smem + 4 * MS;                       // D floats
    float* cld  = hld + D;                             // D floats (contiguous!)

    const int tid  = threadIdx.x;
    const int lane = tid & 31;
    const int wave = tid >> 5;
    const int r0   = blockIdx.x * ROWS_PER_WG;

    // Async-stage this WG's h-side weight rows into LDS once (256 KB).
    // e4 indexes float4s: 4*MS/4 = 16384 of them, 64 per thread.
    for (int e4 = tid; e4 < MS; e4 += blockDim.x) {    // MS == 4*MS/4
        int mat = e4 >> 12;                            // 4096 float4 per matrix
        int e   = (e4 & 4095) * 4;                     // float offset in matrix
        int lr  = e >> 10, k = e & (D - 1);
        const float* Wsrc = (mat == 0) ? Wf : (mat == 1) ? Wi : (mat == 2) ? Wo : WC;
        async_to_lds_b128(wlds + mat * MS + e,
                          Wsrc + (size_t)(r0 + lr) * (2 * D) + k);
    }
    wait_async0();
    __syncthreads();

    const float* Xf = Xp;
    const float* Xi = Xp + (size_t)T * D;
    const float* Xo = Xp + (size_t)2 * T * D;

    unsigned epoch = 0;
    for (int t = 0; t < T; ++t) {
        const float* Hc = Hb + (t & 1) * D;            // c row lives 2*D after
        float* Hn = Hb + ((t + 1) & 1) * D;
        float* Cn = Cb + ((t + 1) & 1) * D;

        __syncthreads();                               // prev-step LDS readers done
#if HAVE_TDM
        if (tid < 32) {                                // wave 0 drives the TDM
            // D# group0: count=1 | lds_addr | global_addr | type=2
            unsigned long long ga = (unsigned long long)(uintptr_t)Hc;
            u32x4 g0;
            g0[0] = 1u;                                // count=1, user mode
            g0[1] = (unsigned)(uintptr_t)hld;          // LDS byte offset
            g0[2] = (unsigned)ga;
            g0[3] = (unsigned)((ga >> 32) & 0x01FFFFFFu) | 0x80000000u; // type=2
            // D# group1: data_size=4B; tensor 1024x2, tile 1024x2, stride 2048.
            i32x8 g1;
            g1[0] = 0x20000;                           // data_size=2 (4 bytes)
            g1[1] = (int)(1024u << 16);                // tensor_dim0[15:0]
            g1[2] = (int)(2u << 16);                   // dim0[31:16]=0 | tensor_dim1=2
            g1[3] = (int)(1024u << 16);                // tile_dim0=1024
            g1[4] = 2;                                 // tile_dim1=2 (h row + c row)
            g1[5] = 2048;                              // tensor_dim0_stride = 2*D
            g1[6] = 0; g1[7] = 0;
            i32x4 z4 = {0, 0, 0, 0};
            i32x8 z8 = {0, 0, 0, 0, 0, 0, 0, 0};
            // clang-23 / therock signature: (g0, g1, g2, g3, g4, cpol)
            __builtin_amdgcn_tensor_load_to_lds(g0, g1, z4, z4, z8, 0);
            __builtin_amdgcn_s_wait_tensorcnt(0);
        }
#else
        {
            const float* Cc = Cb + (t & 1) * D;
            for (int e = tid; e < D; e += blockDim.x) { hld[e] = Hc[e]; cld[e] = Cc[e]; }
        }
#endif
        __syncthreads();

#pragma unroll
        for (int rr = 0; rr < ROWS_PER_WAVE; ++rr) {
            const int lr = wave * ROWS_PER_WAVE + rr;
            const int r  = r0 + lr;
            const float* wcc = WC + (size_t)r * (2 * D) + D;  // c half of W_C row
            float af = 0.f, ai = 0.f, ao = 0.f, ac = 0.f;
            for (int k = lane; k < D; k += 32) {
                float hk = hld[k];
                af = fmaf(wlds[0 * MS + lr * D + k], hk, af);
                ai = fmaf(wlds[1 * MS + lr * D + k], hk, ai);
                ao = fmaf(wlds[2 * MS + lr * D + k], hk, ao);
                ac = fmaf(wlds[3 * MS + lr * D + k], hk, ac);
                ac = fmaf(wcc[k], cld[k], ac);
            }
#pragma unroll
            for (int off = 16; off > 0; off >>= 1) {
                af += __shfl_xor(af, off, 32);
                ai += __shfl_xor(ai, off, 32);
                ao += __shfl_xor(ao, off, 32);
                ac += __shfl_xor(ac, off, 32);
            }
            if (lane == 0) {
                size_t td = (size_t)t * D + r;
                float f  = sigmoidf_(af + bf[r] + Xf[td]);
                float i  = sigmoidf_(ai + bi[r] + Xi[td]);
                float ct = tanhf(ac + bC[r]);          // candidate gate: no x term
                float cn = f * cld[r] + i * ct;
                float o  = sigmoidf_(ao + bo[r] + Xo[td]);
                float hn = o * tanhf(cn);
                out[td] = hn;
                Hn[r]  = hn;
                Cn[r]  = cn;
            }
        }

        // ---- one grid barrier per step ----
        __syncthreads();
        if (tid < 32) {                                // wave 0 only
            __threadfence();                           // publish Hn/Cn device-wide
            if (__builtin_amdgcn_cluster_id_x() != 0) {
                // CDNA5 HW cluster barrier (s_barrier_signal/-3 + s_barrier_wait/-3)
                __builtin_amdgcn_s_cluster_barrier();
            } else if (lane == 0) {
                ++epoch;
                __hip_atomic_fetch_add(bar, 1u, __ATOMIC_ACQ_REL,
                                       __HIP_MEMORY_SCOPE_AGENT);
                const unsigned tgt = epoch * NWG;
                while (__hip_atomic_load(bar, __ATOMIC_ACQUIRE,
                                         __HIP_MEMORY_SCOPE_AGENT) < tgt)
                    __builtin_amdgcn_s_sleep(1);
            }
        }
        __syncthreads();
    }
}

// ---------------------------------------------------------------------------
extern "C" void kernel_launch(void* const* d_in, const int* in_sizes, int n_in,
                              void* d_out, int out_size, void* d_ws, size_t ws_size,
                              hipStream_t stream) {
    (void)in_sizes; (void)n_in; (void)out_size; (void)ws_size;
    const float* X  = (const float*)d_in[0];   // target_seq [T, D]
    const float* h0 = (const float*)d_in[1];
    const float* c0 = (const float*)d_in[2];
    const float* Wf = (const float*)d_in[3];
    const float* bf = (const float*)d_in[4];
    const float* Wi = (const float*)d_in[5];
    const float* bi = (const float*)d_in[6];
    const float* WC = (const float*)d_in[7];
    const float* bC = (const float*)d_in[8];
    const float* Wo = (const float*)d_in[9];
    const float* bo = (const float*)d_in[10];
    float* out = (float*)d_out;

    // workspace layout (needs ~114 MB):
    char* ws = (char*)d_ws;
    float* Wt = (float*)ws;                                          // 12 MB
    float* Xp = (float*)(ws + (size_t)3 * D * D * 4);                // 96 MB
    float* Hb = (float*)(ws + (size_t)3 * D * D * 4 + (size_t)3 * T * D * 4);
    float* Cb = Hb + 2 * D;                                          // contiguous
    unsigned* bar = (unsigned*)(Cb + 2 * D);

    init_state<<<4, 256, 0, stream>>>(h0, c0, Hb, Cb, bar);
    pack_wx<<<(3 * D * D) / 256, 256, 0, stream>>>(Wf, Wi, Wo, Wt);

    dim3 gg(D / 16 / (WAVES * 2), T / 16, 3);                        // (4, 512, 3)
    xproj_wmma<<<gg, 256, (size_t)16 * D * sizeof(float), stream>>>(X, Wt, Xp);

    size_t smem = (size_t)(4 * ROWS_PER_WG * D + 2 * D) * sizeof(float); // 270336 B
    (void)hipFuncSetAttribute((const void*)lstm_scan,
                              hipFuncAttributeMaxDynamicSharedMemorySize,
                              (int)smem);
    lstm_scan<<<NWG, SCAN_THREADS, smem, stream>>>(Wf, Wi, Wo, WC, bf, bi, bC, bo,
                                                   Xp, Hb, Cb, bar, out);
}